// MultiheadStandAttentionNN_53171695125375
// MI455X (gfx1250) — compile-verified
//
#include <hip/hip_runtime.h>
#include <math.h>

// ---------------------------------------------------------------------------
// MultiheadStandAttentionNN for MI455X (gfx1250, wave32, WMMA f32 16x16x4)
//
// Exact algebraic collapse:  heads_out[h,n,p] = x_in[h,n,p] * (A @ Wv_sum)[h,n,p]
//                                             + (A @ Ov)[h,n,p]
// with Wv_sum[h,c,p] = sum_g Wv[h,c,g,p], and softmax logits
// = 2*(x_in . ctr) - ||ctr||^2   (row-constant ||x_in||^2 cancels in softmax).
// All GEMMs run on V_WMMA_F32_16X16X4_F32 (full f32 precision, matches ref).
// ---------------------------------------------------------------------------

typedef __attribute__((ext_vector_type(2))) float v2f;
typedef __attribute__((ext_vector_type(8))) float v8f;

#define NTOK  8192
#define DIN   256
#define DP    64
#define NH    8
#define NC    128
#define DOUT  256
#define HP    512   // NH*DP

// d_ws layout in floats
#define WS_WVS 0        // [8][64][128] (h,p,c)  Wv summed over g, K-major
#define WS_OVT 65536    // [8][64][128] (h,p,c)  Ov transposed, K-major
#define WS_WQT 131072   // [8][64][256] (h,p,i)  Wq transposed, K-major
#define WS_WOT 262144   // [256][512]   (o,r)    Wo transposed, K-major
#define WS_CN2 393216   // [8][128]     ||ctr||^2

static __device__ inline v8f vz8() {
  v8f z = {0.f,0.f,0.f,0.f,0.f,0.f,0.f,0.f};
  return z;
}

static __device__ inline v8f wmma4(v2f a, v2f b, v8f c) {
  // V_WMMA_F32_16X16X4_F32 : D(16x16 f32) = A(16x4 f32) * B(4x16 f32) + C
  return __builtin_amdgcn_wmma_f32_16x16x4_f32(
      false, a, false, b, (short)0, c, false, false);
}

// ---------------------------------------------------------------------------
// Prep: reshuffle weights into K-major layouts + reductions. 512x256 threads.
// ---------------------------------------------------------------------------
__global__ void msann_prep(const float* __restrict__ ctrs,
                           const float* __restrict__ Wv,
                           const float* __restrict__ Ov,
                           const float* __restrict__ Wq,
                           const float* __restrict__ Wo,
                           float* __restrict__ ws) {
  int tid = blockIdx.x * blockDim.x + threadIdx.x;
  if (tid < 65536) {            // Wv_sum (h,p,c) and Ov^T (h,p,c)
    int h = tid >> 13, rem = tid & 8191;
    int p = rem >> 7, c = rem & 127;
    const float* src = Wv + (size_t)h * (NC * DP * DP) + (size_t)c * (DP * DP) + p;
    float s = 0.f;
    #pragma unroll 8
    for (int g = 0; g < DP; ++g) s += src[g * DP];
    ws[WS_WVS + tid] = s;
    ws[WS_OVT + tid] = Ov[(h * NC + c) * DP + p];
  }
  if (tid < 131072) {           // Wq^T (h,p,i) and Wo^T (o,r)
    int h = tid >> 14, rem = tid & 16383;
    int p = rem >> 8, i = rem & 255;
    ws[WS_WQT + tid] = Wq[(i * DP + p) * NH + h];
    int o = tid >> 9, r = tid & 511;
    ws[WS_WOT + tid] = Wo[r * DOUT + o];
  }
  if (tid < 1024) {             // ||ctr||^2
    int h = tid >> 7, c = tid & 127;
    const float* cp = ctrs + (h * NC + c) * DP;
    float s = 0.f;
    #pragma unroll 8
    for (int p = 0; p < DP; ++p) s += cp[p] * cp[p];
    ws[WS_CN2 + tid] = s;
  }
}

// ---------------------------------------------------------------------------
// Fused main kernel: 16 token rows per block, wave w == head w (8 waves).
// Dynamic LDS: x_in [8][16*64] @0 | logits/assign [8][16*128] @8192 |
//              heads_out [16*512] @24576   -> 32768 floats = 128 KB.
// ---------------------------------------------------------------------------
__global__ void msann_fused(const float* __restrict__ x,
                            const float* __restrict__ ctrs,
                            const float* __restrict__ ws,
                            float* __restrict__ out) {
  extern __shared__ float smem[];
  const int lane = threadIdx.x & 31;
  const int h    = threadIdx.x >> 5;   // wave index == head
  const int hi   = lane >> 4;          // half-wave (K-pair select in A/B)
  const int col  = lane & 15;          // row (A) / column (B,C,D) index
  const int n0   = blockIdx.x * 16;

  float* xin_lds = smem + h * (16 * DP);
  float* dist    = smem + 8192 + h * (16 * NC);
  float* hout    = smem + 24576;

  // ---- 1) x_in(16x64) = X(16x256) @ Wq_h(256x64), K=256 ----
  v8f xin[4];
  #pragma unroll
  for (int t = 0; t < 4; ++t) xin[t] = vz8();
  {
    const float* xrow = x + (size_t)(n0 + col) * DIN + 2 * hi;
    const float* wqT  = ws + WS_WQT + h * (DP * DIN);
    for (int kk = 0; kk < 64; ++kk) {
      v2f a = *(const v2f*)(xrow + 4 * kk);
      #pragma unroll
      for (int t = 0; t < 4; ++t) {
        v2f b = *(const v2f*)(wqT + (t * 16 + col) * DIN + 4 * kk + 2 * hi);
        xin[t] = wmma4(a, b, xin[t]);
      }
    }
  }
  // stash x_in row-major in LDS (needed as A-operand for the dots GEMM)
  #pragma unroll
  for (int t = 0; t < 4; ++t)
    #pragma unroll
    for (int j = 0; j < 8; ++j)
      xin_lds[(j + 8 * hi) * DP + t * 16 + col] = xin[t][j];

  // ---- 2) logits(16x128) = 2 * x_in @ ctrs_h^T - ||ctr||^2, K=64 ----
  {
    const float* ch  = ctrs + (size_t)h * NC * DP;      // (c,p): p-contiguous
    const float* cn2 = ws + WS_CN2 + h * NC;
    for (int ct = 0; ct < 8; ++ct) {
      v8f acc = vz8();
      #pragma unroll
      for (int kk = 0; kk < 16; ++kk) {
        v2f a = *(const v2f*)(xin_lds + col * DP + 4 * kk + 2 * hi);
        v2f b = *(const v2f*)(ch + (ct * 16 + col) * DP + 4 * kk + 2 * hi);
        acc = wmma4(a, b, acc);
      }
      float c2 = cn2[ct * 16 + col];
      #pragma unroll
      for (int j = 0; j < 8; ++j)
        dist[(j + 8 * hi) * NC + ct * 16 + col] = 2.f * acc[j] - c2;
    }
  }

  // ---- 3) softmax over c (in-wave; each low-half lane owns one row) ----
  if (hi == 0) {
    float* dr = dist + col * NC;
    float mx = dr[0];
    for (int c = 1; c < NC; ++c) mx = fmaxf(mx, dr[c]);
    float s = 0.f;
    for (int c = 0; c < NC; ++c) { float e = expf(dr[c] - mx); dr[c] = e; s += e; }
    float inv = 1.f / s;
    for (int c = 0; c < NC; ++c) dr[c] *= inv;
  }

  // ---- 4) M = A @ Wv_sum, O = A @ Ov  (16x64 each, K=128) ----
  v8f mac[4], oac[4];
  #pragma unroll
  for (int t = 0; t < 4; ++t) { mac[t] = vz8(); oac[t] = vz8(); }
  {
    const float* wvs = ws + WS_WVS + h * (DP * NC);
    const float* ovT = ws + WS_OVT + h * (DP * NC);
    for (int kk = 0; kk < 32; ++kk) {
      v2f a = *(const v2f*)(dist + col * NC + 4 * kk + 2 * hi);
      #pragma unroll
      for (int t = 0; t < 4; ++t) {
        v2f bm = *(const v2f*)(wvs + (t * 16 + col) * NC + 4 * kk + 2 * hi);
        v2f bo = *(const v2f*)(ovT + (t * 16 + col) * NC + 4 * kk + 2 * hi);
        mac[t] = wmma4(a, bm, mac[t]);
        oac[t] = wmma4(a, bo, oac[t]);
      }
    }
  }

  // heads_out = x_in (.) M + O  — all three share the WMMA C/D layout
  #pragma unroll
  for (int t = 0; t < 4; ++t)
    #pragma unroll
    for (int j = 0; j < 8; ++j)
      hout[(j + 8 * hi) * HP + h * DP + t * 16 + col] =
          xin[t][j] * mac[t][j] + oac[t][j];

  __syncthreads();

  // ---- 5) out(16x256) = heads_out(16x512) @ Wo, wave owns 32 out cols ----
  {
    const float* woT = ws + WS_WOT;     // (o,r): r-contiguous
    const int o0 = h * 32;
    v8f oc[2];
    oc[0] = vz8(); oc[1] = vz8();
    for (int kk = 0; kk < 128; ++kk) {
      v2f a = *(const v2f*)(hout + col * HP + 4 * kk + 2 * hi);
      #pragma unroll
      for (int t = 0; t < 2; ++t) {
        v2f b = *(const v2f*)(woT + (size_t)(o0 + t * 16 + col) * HP + 4 * kk + 2 * hi);
        oc[t] = wmma4(a, b, oc[t]);
      }
    }
    #pragma unroll
    for (int t = 0; t < 2; ++t)
      #pragma unroll
      for (int j = 0; j < 8; ++j)
        out[(size_t)(n0 + j + 8 * hi) * DOUT + o0 + t * 16 + col] = oc[t][j];
  }
}

// ---------------------------------------------------------------------------
extern "C" void kernel_launch(void* const* d_in, const int* in_sizes, int n_in,
                              void* d_out, int out_size, void* d_ws, size_t ws_size,
                              hipStream_t stream) {
  const float* x    = (const float*)d_in[0];
  const float* ctrs = (const float*)d_in[1];
  const float* Wv   = (const float*)d_in[2];
  const float* Ov   = (const float*)d_in[3];
  const float* Wq   = (const float*)d_in[4];
  const float* Wo   = (const float*)d_in[5];
  float* out = (float*)d_out;
  float* ws  = (float*)d_ws;   // needs ~1.6 MB

  msann_prep<<<512, 256, 0, stream>>>(ctrs, Wv, Ov, Wq, Wo, ws);

  const size_t smemBytes = 32768 * sizeof(float);  // 128 KB (< 320 KB/WGP)
  msann_fused<<<NTOK / 16, 256, smemBytes, stream>>>(x, ctrs, ws, out);
}